// SAWN_3959959847661
// MI455X (gfx1250) — compile-verified
//
#include <hip/hip_runtime.h>

#define B_ 4
#define C_ 128
#define H_ 256
#define W_ 256
constexpr int   HW  = H_ * W_;
constexpr float EPS = 1e-5f;

// ---- LDS helpers ----------------------------------------------------------
typedef __attribute__((address_space(3))) float lds_float_t;

static __device__ __forceinline__ unsigned lds_byte_addr(const float* p) {
  // generic (__shared__) pointer -> LDS address space -> 32-bit byte offset
  return (unsigned)(__UINTPTR_TYPE__)(lds_float_t*)(p);
}

// per-lane async copy: 16 bytes global -> LDS, tracked by ASYNCcnt
static __device__ __forceinline__ void async_b128_to_lds(unsigned lds_off,
                                                         const float* gsrc) {
  asm volatile("global_load_async_to_lds_b128 %0, %1, off"
               :: "v"(lds_off), "v"(gsrc)
               : "memory");
}

// ---------------------------------------------------------------------------
// Pass 1: per-(b,c) biased mean/var over a 256x256 plane -> mean, rstd in ws.
// One 256-thread block (8 wave32) per plane. The 256KB plane is streamed
// through LDS with GLOBAL_LOAD_ASYNC_TO_LDS_B128 double buffering (16KB
// chunks, 4 async ops in flight per wave per chunk), reduction reads LDS.
// h reads use default RT policy so the plane stays resident in the 192MB L2
// for pass 2.
// ---------------------------------------------------------------------------
__global__ __launch_bounds__(256) void in_stats_kernel(
    const float* __restrict__ h,
    float* __restrict__ mean_out,
    float* __restrict__ rstd_out) {
  constexpr int CHUNK   = 4096;          // floats per chunk (16 KB)
  constexpr int NCHUNK  = HW / CHUNK;    // 16
  constexpr int ROUNDS  = CHUNK / (256 * 4);  // 4 b128 ops per thread/chunk

  __shared__ float sbuf[2][CHUNK];       // 32 KB double buffer

  const int tid = threadIdx.x;
  const int bc  = blockIdx.x;            // 0 .. B*C-1
  const float* __restrict__ plane = h + (size_t)bc * HW;

  const unsigned lbase0 = lds_byte_addr(&sbuf[0][0]);
  const unsigned lbase1 = lds_byte_addr(&sbuf[1][0]);

  // issue chunk 0 -> buffer 0
#pragma unroll
  for (int r = 0; r < ROUNDS; ++r) {
    const int e = (r * 256 + tid) * 4;   // float index within chunk
    async_b128_to_lds(lbase0 + (unsigned)e * 4u, plane + e);
  }

  float s = 0.f, s2 = 0.f;

  for (int k = 0; k < NCHUNK; ++k) {
    const unsigned cur  = (unsigned)(k & 1);
    const unsigned nxt  = cur ^ 1u;
    const unsigned nbase = nxt ? lbase1 : lbase0;

    if (k + 1 < NCHUNK) {
      const float* gnext = plane + (size_t)(k + 1) * CHUNK;
#pragma unroll
      for (int r = 0; r < ROUNDS; ++r) {
        const int e = (r * 256 + tid) * 4;
        async_b128_to_lds(nbase + (unsigned)e * 4u, gnext + e);
      }
      asm volatile("s_wait_asynccnt 0x4" ::: "memory");  // chunk k landed
    } else {
      asm volatile("s_wait_asynccnt 0x0" ::: "memory");
    }
    __syncthreads();                     // chunk k visible to all waves

    const float4* s4 = reinterpret_cast<const float4*>(&sbuf[cur][0]);
#pragma unroll
    for (int r = 0; r < ROUNDS; ++r) {
      const float4 v = s4[r * 256 + tid];
      s  += (v.x + v.y) + (v.z + v.w);
      s2 += (v.x * v.x + v.y * v.y) + (v.z * v.z + v.w * v.w);
    }
    __syncthreads();                     // buffer cur may be overwritten next
  }

  // wave32 tree reduction
#pragma unroll
  for (int off = 16; off > 0; off >>= 1) {
    s  += __shfl_xor(s,  off, 32);
    s2 += __shfl_xor(s2, off, 32);
  }

  __shared__ float ls[8], ls2[8];
  const int wave = tid >> 5;
  const int lane = tid & 31;
  if (lane == 0) { ls[wave] = s; ls2[wave] = s2; }
  __syncthreads();

  if (tid == 0) {
    float ts = 0.f, ts2 = 0.f;
#pragma unroll
    for (int i = 0; i < 8; ++i) { ts += ls[i]; ts2 += ls2[i]; }
    const float inv  = 1.0f / (float)HW;
    const float mean = ts * inv;
    const float var  = ts2 * inv - mean * mean;   // biased (ddof=0)
    mean_out[bc] = mean;
    rstd_out[bc] = rsqrtf(var + EPS);
  }
}

// ---------------------------------------------------------------------------
// Pass 2: fused bilinear warp(gamma,beta) + mask blend + instance-norm apply.
// Block = (b, row y); thread = column x. Warp geometry computed once per
// pixel, reused across all 128 channels. Channel loop unrolled x4 for MLP.
// out is stored nontemporally and the final h read is nontemporal so the 192MB
// L2 keeps h (from pass 1) and the gamma/beta gather lines resident.
// ---------------------------------------------------------------------------
__global__ __launch_bounds__(256) void warp_fuse_kernel(
    const float* __restrict__ h,     const float* __restrict__ gamma,
    const float* __restrict__ beta,  const float* __restrict__ flow,
    const float* __restrict__ fmask, const float* __restrict__ mean,
    const float* __restrict__ rstd,  float* __restrict__ out) {
  const int x = threadIdx.x;   // 0..255
  const int y = blockIdx.x;    // 0..255
  const int b = blockIdx.y;    // 0..3

  __shared__ float smean[C_], srstd[C_];
  if (x < C_) {
    smean[x] = mean[b * C_ + x];
    srstd[x] = rstd[b * C_ + x];
  }

  const size_t pix   = (size_t)y * W_ + x;
  const size_t fbase = (size_t)b * 2 * HW;
  const float  fx = flow[fbase + pix];
  const float  fy = flow[fbase + HW + pix];
  const float  m  = fmask[(size_t)b * HW + pix];

  // ix = (sx+1)*0.5*(w-1) simplifies to x + fx*(w-1)/w  (same for iy)
  const float ix = (float)x + fx * ((float)(W_ - 1) / (float)W_);
  const float iy = (float)y + fy * ((float)(H_ - 1) / (float)H_);
  const float x0f = floorf(ix), y0f = floorf(iy);
  const float wx = ix - x0f,    wy = iy - y0f;

  const float vx0 = (x0f >= 0.f       && x0f       <= (float)(W_ - 1)) ? 1.f : 0.f;
  const float vx1 = (x0f + 1.f >= 0.f && x0f + 1.f <= (float)(W_ - 1)) ? 1.f : 0.f;
  const float vy0 = (y0f >= 0.f       && y0f       <= (float)(H_ - 1)) ? 1.f : 0.f;
  const float vy1 = (y0f + 1.f >= 0.f && y0f + 1.f <= (float)(H_ - 1)) ? 1.f : 0.f;

  const int x0c = (int)fminf(fmaxf(x0f,       0.f), (float)(W_ - 1));
  const int x1c = (int)fminf(fmaxf(x0f + 1.f, 0.f), (float)(W_ - 1));
  const int y0c = (int)fminf(fmaxf(y0f,       0.f), (float)(H_ - 1));
  const int y1c = (int)fminf(fmaxf(y0f + 1.f, 0.f), (float)(H_ - 1));

  // bilinear weights with validity folded in (OOB corner contributes 0)
  const float w00 = (1.f - wy) * (1.f - wx) * vy0 * vx0;
  const float w01 = (1.f - wy) * wx         * vy0 * vx1;
  const float w10 = wy         * (1.f - wx) * vy1 * vx0;
  const float w11 = wy         * wx         * vy1 * vx1;

  const size_t o00 = (size_t)y0c * W_ + x0c;
  const size_t o01 = (size_t)y0c * W_ + x1c;
  const size_t o10 = (size_t)y1c * W_ + x0c;
  const size_t o11 = (size_t)y1c * W_ + x1c;

  const size_t base = (size_t)b * C_ * HW;
  const float* __restrict__ gp = gamma + base;
  const float* __restrict__ bp = beta  + base;
  const float* __restrict__ hp = h     + base;
  float* __restrict__ op = out + base;
  const float one_m = 1.f - m;

  __syncthreads();

#pragma unroll 4
  for (int c = 0; c < C_; ++c) {
    const size_t pb = (size_t)c * HW;
    if (c + 4 < C_) {  // prefetch 4 channel-planes ahead (global_prefetch_b8)
      const size_t nb = pb + 4 * (size_t)HW;
      __builtin_prefetch(gp + nb + o00, 0, 0);
      __builtin_prefetch(bp + nb + o00, 0, 0);
    }
    const float g00 = gp[pb + o00], g01 = gp[pb + o01];
    const float g10 = gp[pb + o10], g11 = gp[pb + o11];
    const float c00 = bp[pb + o00], c01 = bp[pb + o01];
    const float c10 = bp[pb + o10], c11 = bp[pb + o11];
    const float hv  = __builtin_nontemporal_load(hp + pb + pix); // last use of h

    const float gw = w00 * g00 + w01 * g01 + w10 * g10 + w11 * g11;
    const float bw = w00 * c00 + w01 * c01 + w10 * c10 + w11 * c11;
    const float hn = (hv - smean[c]) * srstd[c];

    __builtin_nontemporal_store((gw * m + hv * one_m) * hn + bw,
                                op + pb + pix);   // write-once stream
  }
}

// ---------------------------------------------------------------------------
extern "C" void kernel_launch(void* const* d_in, const int* in_sizes, int n_in,
                              void* d_out, int out_size, void* d_ws, size_t ws_size,
                              hipStream_t stream) {
  (void)in_sizes; (void)n_in; (void)out_size; (void)ws_size;

  const float* h     = (const float*)d_in[0];
  const float* gamma = (const float*)d_in[1];
  const float* beta  = (const float*)d_in[2];
  const float* flow  = (const float*)d_in[3];
  const float* fmask = (const float*)d_in[4];
  float*       out   = (float*)d_out;

  float* mean = (float*)d_ws;        // B*C = 512 floats
  float* rstd = mean + B_ * C_;      // next 512 floats

  in_stats_kernel<<<B_ * C_, 256, 0, stream>>>(h, mean, rstd);

  dim3 grid(H_, B_);
  warp_fuse_kernel<<<grid, 256, 0, stream>>>(h, gamma, beta, flow, fmask,
                                             mean, rstd, out);
}